// LASA_31799937859696
// MI455X (gfx1250) — compile-verified
//
#include <hip/hip_runtime.h>
#include <hip/hip_bf16.h>
#include <math.h>

typedef __attribute__((ext_vector_type(2))) float v2f;
typedef __attribute__((ext_vector_type(4))) float v4f;
typedef __attribute__((ext_vector_type(8))) float v8f;

#define Bn   4
#define Cn   256
#define Hn   128
#define Wn   128
#define GCn  64
#define NPIX (Bn*Hn*Wn)   // 65536
#define HW   (Hn*Wn)      // 16384

// shared memory layout for the e-kernel (floats)
#define SM_G    0            // 4096
#define SM_QR   4096         // 64
#define SM_X    4160         // 4*11*128 = 5632 (max), 16B-aligned base
#define SM_S    9792         // 512
#define SM_TOT  10304        // ~41 KB

// ---------------- precompute kernels ----------------

// G[g][c][c2] = sum_o qw[g][o][c] * kw[g][o][c2]
__global__ __launch_bounds__(64) void lasa_precompG(const float* __restrict__ qw,
                                                    const float* __restrict__ kw,
                                                    float* __restrict__ G) {
    int g = blockIdx.y, c = blockIdx.x, c2 = threadIdx.x;
    const float* qg = qw + g * 4096;
    const float* kg = kw + g * 4096;
    float s = 0.f;
    for (int o = 0; o < 64; ++o) s += qg[o * 64 + c] * kg[o * 64 + c2];
    G[g * 4096 + c * 64 + c2] = s;
}

// qR[g][c] = sum_o qw[g][o][c] * (sum_r rk_g[o][r])
__global__ __launch_bounds__(64) void lasa_precompQR(const float* __restrict__ qw,
                                                     const float* __restrict__ rk0,
                                                     const float* __restrict__ rk1,
                                                     const float* __restrict__ rk2,
                                                     const float* __restrict__ rk3,
                                                     float* __restrict__ qR) {
    int g = blockIdx.x;
    const float* rk = (g == 0) ? rk0 : (g == 1) ? rk1 : (g == 2) ? rk2 : rk3;
    int nr = 2 * (5 + 2 * g) - 1;   // 9,13,17,21
    __shared__ float R[64];
    int t = threadIdx.x;
    float s = 0.f;
    for (int r = 0; r < nr; ++r) s += rk[t * nr + r];
    R[t] = s;
    __syncthreads();
    float q = 0.f;
    for (int o = 0; o < 64; ++o) q += qw[g * 4096 + o * 64 + t] * R[o];
    qR[g * 64 + t] = q;
}

// Wg[o][g] = sum_{c in group g} fusion_w[o][c]
__global__ __launch_bounds__(256) void lasa_precompWg(const float* __restrict__ fw,
                                                      float* __restrict__ Wg) {
    int o = threadIdx.x;
    for (int gg = 0; gg < 4; ++gg) {
        float s = 0.f;
        for (int c = 0; c < 64; ++c) s += fw[o * 256 + gg * 64 + c];
        Wg[o * 4 + gg] = s;
    }
}

// ---------------- main kernel: e via WMMA, PAD compile-time ----------------
// e(p) = sum_c (x_g+r_q)[c,p] * ( (G_g @ S_xg)[c,p] + qR_g[c] )
template <int PAD>
__device__ __forceinline__ void lasa_e_impl(float* __restrict__ sm,
                                            const float* __restrict__ x,
                                            const float* __restrict__ rq,
                                            const float* __restrict__ Gw,
                                            const float* __restrict__ qRw,
                                            float* __restrict__ e,
                                            int h, int b, int g) {
    constexpr int ROWS   = 2 * PAD + 1;      // 5,7,9,11
    constexpr int CH_ROW = ROWS * 32;        // float4 chunks per channel
    constexpr int CHUNKS = 4 * CH_ROW;       // float4 chunks per K-step
    constexpr int NCH    = (CHUNKS + 255) / 256;  // per-thread chunks (3..6)

    float* Gs  = sm + SM_G;
    float* qRs = sm + SM_QR;
    float* Xs  = sm + SM_X;
    float* Ss  = sm + SM_S;

    const int tid = threadIdx.x;
    for (int i = tid; i < 4096; i += 256) Gs[i] = Gw[g * 4096 + i];
    if (tid < 64) qRs[tid] = qRw[g * 64 + tid];

    const int lane = tid & 31;
    const int wave = tid >> 5;
    const int nlo  = lane & 15;
    const int half = lane >> 4;
    const int w    = wave * 16 + nlo;

    v8f acc[4];
#pragma unroll
    for (int m = 0; m < 4; ++m) acc[m] = (v8f){0,0,0,0,0,0,0,0};

    const long xgbase = (long)(b * Cn + g * GCn) * HW;

    for (int kc = 0; kc < 16; ++kc) {
        __syncthreads();                    // Ss consumers of prev iter done
        // ---- stage x tile (4 channels, ROWS rows with halo) via b128 ----
        v4f vals[NCH];
#pragma unroll
        for (int t = 0; t < NCH; ++t) {     // issue all global loads first
            int idx = tid + t * 256;
            v4f v = (v4f){0.f, 0.f, 0.f, 0.f};
            if (idx < CHUNKS) {
                int cc  = idx / CH_ROW;     // constant divisor
                int rem = idx - cc * CH_ROW;
                int r   = rem >> 5;
                int wq  = rem & 31;
                int hh  = h - PAD + r;
                if (hh >= 0 && hh < Hn)
                    v = *(const v4f*)(x + xgbase + (long)(kc * 4 + cc) * HW
                                        + hh * Wn + wq * 4);
            }
            vals[t] = v;
        }
#pragma unroll
        for (int t = 0; t < NCH; ++t) {     // then all LDS stores
            int idx = tid + t * 256;
            if (idx < CHUNKS) {
                int cc  = idx / CH_ROW;
                int rem = idx - cc * CH_ROW;
                int r   = rem >> 5;
                int wq  = rem & 31;
                *(v4f*)(Xs + (cc * ROWS + r) * 128 + wq * 4) = vals[t];
            }
        }
        __syncthreads();
        // ---- axial sums into Ss ----
        for (int i = tid; i < 512; i += 256) {
            int cc = i >> 7;
            int ww = i & 127;
            float s = 0.f;
#pragma unroll
            for (int r = 0; r < ROWS; ++r) s += Xs[(cc * ROWS + r) * 128 + ww];
            const float* crow = &Xs[(cc * ROWS + PAD) * 128];
#pragma unroll
            for (int dx = 1; dx <= PAD; ++dx) {
                if (ww - dx >= 0) s += crow[ww - dx];
                if (ww + dx < Wn) s += crow[ww + dx];
            }
            Ss[cc * 128 + ww] = s;
        }
        __syncthreads();
        // ---- WMMA K-step: B fragment 4x16 (KxN), row K = vgpr + 2*laneHalf
        v2f bf;
        bf.x = Ss[(half * 2 + 0) * 128 + w];
        bf.y = Ss[(half * 2 + 1) * 128 + w];
#pragma unroll
        for (int m = 0; m < 4; ++m) {
            // A fragment: 16x4 (MxK), M = lane&15, K = kc*4 + half*2 + {0,1}
            v2f af;
            af.x = Gs[(m * 16 + nlo) * 64 + kc * 4 + half * 2 + 0];
            af.y = Gs[(m * 16 + nlo) * 64 + kc * 4 + half * 2 + 1];
            acc[m] = __builtin_amdgcn_wmma_f32_16x16x4_f32(
                false, af, false, bf, (short)0, acc[m], false, false);
        }
    }

    // epilogue: D vgpr j holds channel m*16 + j + 8*half, column w
    float partial = 0.f;
#pragma unroll
    for (int m = 0; m < 4; ++m) {
#pragma unroll
        for (int j = 0; j < 8; ++j) {
            int cl = m * 16 + j + half * 8;
            float qv = x[xgbase + (long)cl * HW + h * Wn + w] + rq[g * GCn + cl];
            partial += (acc[m][j] + qRs[cl]) * qv;
        }
    }
    partial += __shfl_xor(partial, 16);
    if (half == 0)
        e[((g * Bn + b) * Hn + h) * Wn + w] = partial;
}

__global__ __launch_bounds__(256) void lasa_e_kernel(const float* __restrict__ x,
                                                     const float* __restrict__ rq,
                                                     const float* __restrict__ Gw,
                                                     const float* __restrict__ qRw,
                                                     float* __restrict__ e) {
    __shared__ float sm[SM_TOT];
    const int h = blockIdx.x;
    const int b = blockIdx.y;
    const int g = blockIdx.z;           // block-uniform switch: no divergence
    switch (g) {
        case 0: lasa_e_impl<2>(sm, x, rq, Gw, qRw, e, h, b, 0); break;
        case 1: lasa_e_impl<3>(sm, x, rq, Gw, qRw, e, h, b, 1); break;
        case 2: lasa_e_impl<4>(sm, x, rq, Gw, qRw, e, h, b, 2); break;
        default: lasa_e_impl<5>(sm, x, rq, Gw, qRw, e, h, b, 3); break;
    }
}

// ---------------- stats: deterministic 2-stage reduction ----------------

__global__ __launch_bounds__(256) void lasa_stats1(const float* __restrict__ e,
                                                   float* __restrict__ part) {
    __shared__ float red[256 * 14];
    float s[14];
#pragma unroll
    for (int i = 0; i < 14; ++i) s[i] = 0.f;
    int base = blockIdx.x * 512 + threadIdx.x;
    for (int it = 0; it < 2; ++it) {
        int p = base + it * 256;
        float e0 = e[p], e1 = e[NPIX + p], e2 = e[2 * NPIX + p], e3 = e[3 * NPIX + p];
        s[0] += e0; s[1] += e1; s[2] += e2; s[3] += e3;
        s[4] += e0 * e0; s[5] += e0 * e1; s[6]  += e0 * e2; s[7]  += e0 * e3;
        s[8] += e1 * e1; s[9] += e1 * e2; s[10] += e1 * e3;
        s[11] += e2 * e2; s[12] += e2 * e3; s[13] += e3 * e3;
    }
#pragma unroll
    for (int i = 0; i < 14; ++i) red[i * 256 + threadIdx.x] = s[i];
    __syncthreads();
    for (int stride = 128; stride > 0; stride >>= 1) {
        if (threadIdx.x < stride)
            for (int i = 0; i < 14; ++i)
                red[i * 256 + threadIdx.x] += red[i * 256 + threadIdx.x + stride];
        __syncthreads();
    }
    if (threadIdx.x < 14) part[blockIdx.x * 14 + threadIdx.x] = red[threadIdx.x * 256];
}

__global__ __launch_bounds__(256) void lasa_stats2(const float* __restrict__ part,
                                                   const float* __restrict__ Wg,
                                                   const float* __restrict__ gamma,
                                                   const float* __restrict__ beta,
                                                   float* __restrict__ aw,
                                                   float* __restrict__ shift) {
    __shared__ float st[14];
    if (threadIdx.x < 14) {
        float s = 0.f;
        for (int bb = 0; bb < 128; ++bb) s += part[bb * 14 + threadIdx.x];
        st[threadIdx.x] = s * (1.0f / (float)NPIX);
    }
    __syncthreads();
    int o = threadIdx.x;
    float m0 = st[0], m1 = st[1], m2 = st[2], m3 = st[3];
    float M[4][4];
    M[0][0] = st[4];  M[0][1] = st[5];  M[0][2] = st[6];  M[0][3] = st[7];
    M[1][1] = st[8];  M[1][2] = st[9];  M[1][3] = st[10];
    M[2][2] = st[11]; M[2][3] = st[12]; M[3][3] = st[13];
    M[1][0] = M[0][1]; M[2][0] = M[0][2]; M[3][0] = M[0][3];
    M[2][1] = M[1][2]; M[3][1] = M[1][3]; M[3][2] = M[2][3];
    float wv[4];
#pragma unroll
    for (int gg = 0; gg < 4; ++gg) wv[gg] = Wg[o * 4 + gg];
    float mean = wv[0] * m0 + wv[1] * m1 + wv[2] * m2 + wv[3] * m3;
    float e2 = 0.f;
#pragma unroll
    for (int gg = 0; gg < 4; ++gg)
#pragma unroll
        for (int g2 = 0; g2 < 4; ++g2) e2 += wv[gg] * wv[g2] * M[gg][g2];
    float var = e2 - mean * mean;
    float sc  = gamma[o] * rsqrtf(var + 1e-5f);
    float sh  = beta[o] - mean * sc;
#pragma unroll
    for (int gg = 0; gg < 4; ++gg) aw[o * 4 + gg] = sc * wv[gg];
    shift[o] = sh;
}

// ---------------- final fused elementwise kernel ----------------

__global__ __launch_bounds__(128) void lasa_final(const float* __restrict__ x,
                                                  const float* __restrict__ e,
                                                  const float* __restrict__ aw,
                                                  const float* __restrict__ shift,
                                                  float* __restrict__ out) {
    int w = threadIdx.x;
    int h = blockIdx.x;
    int b = blockIdx.y;
    int oc0 = blockIdx.z * 32;
    int pix = (b * Hn + h) * Wn + w;
    float e0 = e[pix], e1 = e[NPIX + pix], e2 = e[2 * NPIX + pix], e3 = e[3 * NPIX + pix];
    for (int oo = 0; oo < 32; ++oo) {
        int o = oc0 + oo;
        float z = shift[o] + aw[o * 4 + 0] * e0 + aw[o * 4 + 1] * e1
                           + aw[o * 4 + 2] * e2 + aw[o * 4 + 3] * e3;
        float act = (z <= 0.f) ? 0.5f : 1.f / (1.f + __expf(-z));
        long idx = ((long)(b * Cn + o) * Hn + h) * Wn + w;
        out[idx] = x[idx] * act;
    }
}

// ---------------- launch ----------------

extern "C" void kernel_launch(void* const* d_in, const int* in_sizes, int n_in,
                              void* d_out, int out_size, void* d_ws, size_t ws_size,
                              hipStream_t stream) {
    const float* x    = (const float*)d_in[0];
    const float* qw   = (const float*)d_in[1];
    const float* kw   = (const float*)d_in[2];
    const float* r_q  = (const float*)d_in[3];
    const float* rk0  = (const float*)d_in[4];
    const float* rk1  = (const float*)d_in[5];
    const float* rk2  = (const float*)d_in[6];
    const float* rk3  = (const float*)d_in[7];
    const float* fw   = (const float*)d_in[8];
    const float* gam  = (const float*)d_in[9];
    const float* bet  = (const float*)d_in[10];
    float* out = (float*)d_out;
    float* ws  = (float*)d_ws;

    // workspace layout (floats)
    float* wG     = ws;                 // 4*4096 = 16384
    float* wQR    = ws + 16384;         // 256
    float* wWg    = ws + 16640;         // 1024
    float* wE     = ws + 17664;         // 4*NPIX = 262144
    float* wPart  = ws + 279808;        // 128*14 = 1792
    float* wA     = ws + 281600;        // 1024
    float* wShift = ws + 282624;        // 256   (end: 282880 floats ~ 1.1 MB)

    lasa_precompG<<<dim3(64, 4), 64, 0, stream>>>(qw, kw, wG);
    lasa_precompQR<<<4, 64, 0, stream>>>(qw, rk0, rk1, rk2, rk3, wQR);
    lasa_precompWg<<<1, 256, 0, stream>>>(fw, wWg);

    lasa_e_kernel<<<dim3(Hn, Bn, 4), 256, 0, stream>>>(x, r_q, wG, wQR, wE);

    lasa_stats1<<<128, 256, 0, stream>>>(wE, wPart);
    lasa_stats2<<<1, 256, 0, stream>>>(wPart, wWg, gam, bet, wA, wShift);

    lasa_final<<<dim3(Hn, Bn, 8), 128, 0, stream>>>(x, wE, wA, wShift, out);
}